// NoCarryOverSRlayer_80805514707488
// MI455X (gfx1250) — compile-verified
//
#include <hip/hip_runtime.h>

#define NUM_HEAD 8
#define DIM_HEAD 32
#define IN_DIM   256   // NUM_HEAD * DIM_HEAD
#define SLEN     1024
#define BSZ      8

typedef float v2f __attribute__((ext_vector_type(2)));
typedef float v8f __attribute__((ext_vector_type(8)));

// ---------------- wave32 cross-lane helpers (gfx1250 is wave32-only) --------
// Uniform broadcast of lane `l` (compile-time constant after unroll):
// lowers to v_readlane_b32 sN, v, l -> scalar operand of v_fmac. No DS wait.
__device__ __forceinline__ float bcast(float v, int l) {
  return __builtin_bit_cast(
      float, __builtin_amdgcn_readlane(__builtin_bit_cast(int, v), l));
}
__device__ __forceinline__ float wave_sum(float v) {
#pragma unroll
  for (int off = 16; off > 0; off >>= 1) v += __shfl_xor(v, off, 32);
  return v;  // full result in every lane
}
__device__ __forceinline__ float wave_max(float v) {
#pragma unroll
  for (int off = 16; off > 0; off >>= 1) v = fmaxf(v, __shfl_xor(v, off, 32));
  return v;
}
__device__ __forceinline__ float wave_softmax(float v) {
  float m = wave_max(v);
  float e = __expf(v - m);
  float s = wave_sum(e);
  return e / s;
}
__device__ __forceinline__ float sigmoidf(float x) {
  return 1.0f / (1.0f + __expf(-x));
}

// ---------------------------------------------------------------------------
// Kernel 1: sequential SRWM scan. One wave32 per (batch, head) pair.
// Lane m owns row m of Wy/Wq/Wk (32 f32 each, kept in VGPRs).
// w_b is kept TRANSPOSED: lane t holds column t&3 of w_b (32 VGPRs), so the
// beta / delta-beta dot products ride along inside the broadcast mat-vec
// loops as one extra v_fmac per iteration — no cross-lane reduction trees.
// Only the 3 softmaxes still use ds_swizzle reductions (unavoidable).
// ---------------------------------------------------------------------------
__global__ __launch_bounds__(32, 1)
void srwm_scan_kernel(const float* __restrict__ hin,   // (S,B,256)
                      const float* __restrict__ Wy0,   // (1,H,32,32)
                      const float* __restrict__ Wq0,
                      const float* __restrict__ Wk0,
                      const float* __restrict__ wb0,   // (1,H,32,4)
                      float* __restrict__ ys)          // (S,B,256) scratch
{
  const int bh = blockIdx.x;      // 0..63
  const int b  = bh >> 3;
  const int hd = bh & 7;
  const int m  = threadIdx.x;     // row index / element index within head

  // ---- load per-head initial state (broadcast over batch) ----
  float Wy[32], Wq[32], Wk[32];
  {
    const float* py = Wy0 + (hd * 32 + m) * 32;
    const float* pq = Wq0 + (hd * 32 + m) * 32;
    const float* pk = Wk0 + (hd * 32 + m) * 32;
#pragma unroll
    for (int j = 0; j < 32; ++j) { Wy[j] = py[j]; Wq[j] = pq[j]; Wk[j] = pk[j]; }
  }
  // Transposed w_b: lane l holds wb[:, l&3] (lanes 4..31 = duplicates, kept
  // consistent by identical updates; broadcasts always read lanes 0..3).
  float wbT[32];
  {
    const int t = m & 3;
    const float* pb = wb0 + hd * (32 * 4) + t;
#pragma unroll
    for (int j = 0; j < 32; ++j) wbT[j] = pb[j * 4];
  }

  const size_t row_stride = (size_t)BSZ * IN_DIM;
  const float* xptr = hin + b * IN_DIM + hd * DIM_HEAD + m;
  float*       yptr = ys  + b * IN_DIM + hd * DIM_HEAD + m;

  float xcur = xptr[0];

#pragma unroll 1
  for (int s = 0; s < SLEN; ++s) {
    // x_t = softmax(h_t) over the head dimension (lanes)
    float vx = wave_softmax(xcur);

    // software pipeline: issue next step's x load now (clamped at the end)
    const int sn = (s + 1 < SLEN) ? (s + 1) : s;
    float xnext = xptr[(size_t)sn * row_stride];

    // y = Wy x ; q' = Wq x ; k' = Wk x ; bacc = wb^T x (lane t -> gate t)
    float y = 0.f, q = 0.f, k = 0.f, bacc = 0.f;
#pragma unroll
    for (int j = 0; j < 32; ++j) {
      float xj = bcast(vx, j);
      y    = fmaf(Wy[j],  xj, y);
      q    = fmaf(Wq[j],  xj, q);
      k    = fmaf(Wk[j],  xj, k);
      bacc = fmaf(wbT[j], xj, bacc);
    }
    float qs = wave_softmax(q);
    float ks = wave_softmax(k);

    // beta_t = sigmoid(wb^T x): one sigmoid covers all 4 gates (lane t),
    // then 4 readlanes give wave-uniform scalars.
    float sig = sigmoidf(bacc);
    float b0 = bcast(sig, 0);
    float b1 = bcast(sig, 1);
    float b2 = bcast(sig, 2);
    float b3 = bcast(sig, 3);

    // u = q - k ; d* = W*(u) ; dbacc = wb^T u (lane t -> db_t)
    float u = qs - ks;
    float dy = 0.f, dq = 0.f, dk = 0.f, dbacc = 0.f;
#pragma unroll
    for (int j = 0; j < 32; ++j) {
      float uj = bcast(u, j);
      dy    = fmaf(Wy[j],  uj, dy);
      dq    = fmaf(Wq[j],  uj, dq);
      dk    = fmaf(Wk[j],  uj, dk);
      dbacc = fmaf(wbT[j], uj, dbacc);
    }

    // rank-1 delta updates: W += beta * d (x) k ; wbT[:,t] += b3*db_t*k
    float cy = b0 * dy, cq = b1 * dq, ck = b2 * dk;
    float cb = b3 * dbacc;  // per-lane: b3 * db_{lane&3}
#pragma unroll
    for (int j = 0; j < 32; ++j) {
      float kj = bcast(ks, j);
      Wy[j]  = fmaf(cy, kj, Wy[j]);
      Wq[j]  = fmaf(cq, kj, Wq[j]);
      Wk[j]  = fmaf(ck, kj, Wk[j]);
      wbT[j] = fmaf(cb, kj, wbT[j]);
    }

    // emit y_t (pre-update Wy read, as in the reference)
    yptr[(size_t)s * row_stride] = y;
    xcur = xnext;
  }
}

// ---------------------------------------------------------------------------
// Kernel 2: out = h + ys @ W_out^T via V_WMMA_F32_16X16X4_F32.
// One wave per 16x16 output tile; K=256 stepped by 4.
// A frag (16x4 f32): lane<16 -> A[m][K0,K1]; lane>=16 -> A[m][K2,K3].
// B frag (4x16 f32): transposed mirror; B[k][n] = Wout[tile_n*16+n][k],
//   so each lane's B frag is also 2 contiguous floats from a Wout row.
// C/D (16x16 f32, 8 VGPRs): row = i + 8*(lane>=16), col = lane&15.
// ---------------------------------------------------------------------------
__global__ __launch_bounds__(256)
void srwm_outproj_wmma(const float* __restrict__ ys,   // (8192,256)
                       const float* __restrict__ Wout, // (256,256) row-major
                       const float* __restrict__ hin,  // (8192,256)
                       float* __restrict__ out)        // (8192,256)
{
  const int lane = threadIdx.x & 31;
  const int wave = threadIdx.x >> 5;
  const int tile = blockIdx.x * 8 + wave;  // 0..8191
  const int tm = tile >> 4;                // 0..511  (M tiles of 16)
  const int tn = tile & 15;                // 0..15   (N tiles of 16)
  const int hi = lane >> 4;                // 0: K pair {0,1}, 1: {2,3}
  const int lm = lane & 15;

  const float* arow = ys   + (size_t)(tm * 16 + lm) * IN_DIM;
  const float* brow = Wout + (size_t)(tn * 16 + lm) * IN_DIM;

  v8f acc = {};
#pragma unroll 4
  for (int kb = 0; kb < IN_DIM; kb += 4) {
    const float2 af = reinterpret_cast<const float2*>(arow + kb)[hi];
    const float2 bf = reinterpret_cast<const float2*>(brow + kb)[hi];
    v2f a;  a[0] = af.x;  a[1] = af.y;
    v2f bb; bb[0] = bf.x; bb[1] = bf.y;
    // 8 args: (neg_a, A, neg_b, B, c_mod, C, reuse_a, reuse_b)
    acc = __builtin_amdgcn_wmma_f32_16x16x4_f32(
        false, a, false, bb, (short)0, acc, false, false);
  }

  // epilogue: residual add + store
#pragma unroll
  for (int i = 0; i < 8; ++i) {
    const int row = tm * 16 + i + 8 * hi;
    const int col = tn * 16 + lm;
    const size_t idx = (size_t)row * IN_DIM + col;
    out[idx] = hin[idx] + acc[i];
  }
}

// ---------------------------------------------------------------------------
extern "C" void kernel_launch(void* const* d_in, const int* in_sizes, int n_in,
                              void* d_out, int out_size, void* d_ws, size_t ws_size,
                              hipStream_t stream) {
  const float* h    = (const float*)d_in[0];  // (S,B,256)
  const float* W_y  = (const float*)d_in[1];  // (1,H,32,32)
  const float* W_q  = (const float*)d_in[2];
  const float* W_k  = (const float*)d_in[3];
  const float* w_b  = (const float*)d_in[4];  // (1,H,32,4)
  const float* Wout = (const float*)d_in[5];  // (256,256)
  float* out = (float*)d_out;
  float* ys  = (float*)d_ws;                  // needs S*B*256*4 = 8 MB

  // Phase 1: sequential scan, 64 independent (b,h) chains, one wave32 each.
  srwm_scan_kernel<<<BSZ * NUM_HEAD, 32, 0, stream>>>(h, W_y, W_q, W_k, w_b, ys);

  // Phase 2: out = h + ys @ Wout^T, WMMA f32 tiles. 8192 tiles / 8 waves per block.
  srwm_outproj_wmma<<<(SLEN * BSZ / 16) * (IN_DIM / 16) / 8, 256, 0, stream>>>(
      ys, Wout, h, out);
}